// GAT_37744172597909
// MI455X (gfx1250) — compile-verified
//
#include <hip/hip_runtime.h>
#include <hip/hip_bf16.h>
#include <math.h>

#define IN_DIM   128
#define HID      64
#define HEADS    2
#define F        (HEADS * HID)   // 128 = feature width of every layer
#define NEG_SLOPE 0.2f

typedef __attribute__((ext_vector_type(2))) float v2f;
typedef __attribute__((ext_vector_type(8))) float v8f;

// ---------------------------------------------------------------------------
// GEMM: C[M x 128] = A[M x 128] @ B[128 x 128], fp32 WMMA 16x16x4.
// One wave per 16x32 output strip (two 16x16 tiles sharing the A fragment):
// halves A-fragment traffic and gives two independent WMMA chains per wave.
// Fragment layout per CDNA5 ISA 7.12.2 (fp32): K = vgpr_idx + 2*(lane>>4).
// ---------------------------------------------------------------------------
__global__ void k_gemm128(const float* __restrict__ A, const float* __restrict__ B,
                          float* __restrict__ C, int M, int Mtiles)
{
    const int wid  = blockIdx.x * (blockDim.x >> 5) + (threadIdx.x >> 5);
    const int lane = threadIdx.x & 31;
    const int tm = wid >> 2;                // 4 strips of 32 columns
    const int tg = wid & 3;
    if (tm >= Mtiles) return;               // uniform per-wave exit (EXEC stays full)

    const int half = lane >> 4;             // 0 or 1
    const int ln   = lane & 15;
    int ar = tm * 16 + ln;
    if (ar >= M) ar = M - 1;                // clamp: keep EXEC all-ones for WMMA
    const float* __restrict__ Arow = A + (size_t)ar * IN_DIM;
    const float* __restrict__ B0   = B + tg * 32 + ln;
    const float* __restrict__ B1   = B0 + 16;

    v8f acc0 = {}, acc1 = {};
    #pragma unroll 4
    for (int k = 0; k < IN_DIM; k += 4) {
        const int k0 = k + 2 * half;        // K for vgpr 0; vgpr 1 -> k0+1
        v2f a, b0, b1;
        a.x  = Arow[k0];
        a.y  = Arow[k0 + 1];
        b0.x = B0[(size_t)k0 * F];
        b0.y = B0[(size_t)(k0 + 1) * F];
        b1.x = B1[(size_t)k0 * F];
        b1.y = B1[(size_t)(k0 + 1) * F];
        acc0 = __builtin_amdgcn_wmma_f32_16x16x4_f32(
                   false, a, false, b0, (short)0, acc0, false, false);
        acc1 = __builtin_amdgcn_wmma_f32_16x16x4_f32(
                   false, a, false, b1, (short)0, acc1, false, false);
    }

    float* __restrict__ C0 = C + tg * 32 + ln;
    float* __restrict__ C1 = C0 + 16;
    if (tm * 16 + 16 <= M) {                // wave-uniform: branch-free common path
        #pragma unroll
        for (int vg = 0; vg < 8; ++vg) {
            const size_t m = (size_t)(tm * 16 + vg + 8 * half);
            C0[m * F] = acc0[vg];
            C1[m * F] = acc1[vg];
        }
    } else {                                // single tail tile: guarded stores
        #pragma unroll
        for (int vg = 0; vg < 8; ++vg) {
            const int m = tm * 16 + vg + 8 * half;
            if (m < M) {
                C0[(size_t)m * F] = acc0[vg];
                C1[(size_t)m * F] = acc1[vg];
            }
        }
    }
}

// ---------------------------------------------------------------------------
// Per-(node, head) attention coefficients: as = <h, a_src>, ad = <h, a_dst>
// ---------------------------------------------------------------------------
__global__ void k_alpha(const float* __restrict__ h, const float* __restrict__ a_src,
                        const float* __restrict__ a_dst, float* __restrict__ as_,
                        float* __restrict__ ad_, int n)
{
    const int tid = blockIdx.x * blockDim.x + threadIdx.x;
    if (tid >= n * HEADS) return;
    const int node = tid >> 1, head = tid & 1;
    const float4* __restrict__ hv = (const float4*)(h + (size_t)node * F + head * HID);
    const float4* __restrict__ av = (const float4*)(a_src + head * HID);
    const float4* __restrict__ dv = (const float4*)(a_dst + head * HID);
    float s = 0.f, d = 0.f;
    #pragma unroll
    for (int i = 0; i < HID / 4; ++i) {
        const float4 hh = hv[i], aa = av[i], dd = dv[i];
        s += hh.x * aa.x + hh.y * aa.y + hh.z * aa.z + hh.w * aa.w;
        d += hh.x * dd.x + hh.y * dd.y + hh.z * dd.z + hh.w * dd.w;
    }
    as_[tid] = s;
    ad_[tid] = d;
}

// ---------------------------------------------------------------------------
// Zero accumulators / init segment-max to -FLT_MAX, segment-sum to 0
// ---------------------------------------------------------------------------
__global__ void k_init(float* __restrict__ agg, float* __restrict__ m_,
                       float* __restrict__ s_, int n)
{
    const int tid = blockIdx.x * blockDim.x + threadIdx.x;
    if (tid < n * F) agg[tid] = 0.f;
    if (tid < n * HEADS) { m_[tid] = -3.402823466e38f; s_[tid] = 0.f; }
}

__device__ __forceinline__ void atomicMaxF(float* addr, float v)
{
    // Classic sign-split trick: correct ordered max on raw float bits.
    if (v >= 0.f) atomicMax((int*)addr, __float_as_int(v));
    else          atomicMin((unsigned int*)addr, __float_as_uint(v));
}

__device__ __forceinline__ void edge_endpoints(const int* __restrict__ ei,
                                               int e, int E, int& src, int& dst)
{
    if (e < E) { src = ei[e]; dst = ei[E + e]; }
    else       { src = dst = e - E; }        // appended self-loops
}

__device__ __forceinline__ float leaky(float v)
{
    return v > 0.f ? v : NEG_SLOPE * v;
}

// ---------------------------------------------------------------------------
// Edge pass 1: segment max of leaky_relu(as[src] + ad[dst]) over dst
// ---------------------------------------------------------------------------
__global__ void k_edge_max(const int* __restrict__ ei, const float* __restrict__ as_,
                           const float* __restrict__ ad_, float* __restrict__ m_,
                           int E, int Etot)
{
    const int tid = blockIdx.x * blockDim.x + threadIdx.x;
    if (tid >= Etot * HEADS) return;
    const int e = tid >> 1, head = tid & 1;
    int src, dst;
    edge_endpoints(ei, e, E, src, dst);
    const float v = leaky(as_[src * HEADS + head] + ad_[dst * HEADS + head]);
    atomicMaxF(&m_[dst * HEADS + head], v);
}

// ---------------------------------------------------------------------------
// Edge pass 2: segment sum of exp(e - m[dst])
// ---------------------------------------------------------------------------
__global__ void k_edge_sum(const int* __restrict__ ei, const float* __restrict__ as_,
                           const float* __restrict__ ad_, const float* __restrict__ m_,
                           float* __restrict__ s_, int E, int Etot)
{
    const int tid = blockIdx.x * blockDim.x + threadIdx.x;
    if (tid >= Etot * HEADS) return;
    const int e = tid >> 1, head = tid & 1;
    int src, dst;
    edge_endpoints(ei, e, E, src, dst);
    const float v = leaky(as_[src * HEADS + head] + ad_[dst * HEADS + head]);
    atomicAdd(&s_[dst * HEADS + head], __expf(v - m_[dst * HEADS + head]));
}

// ---------------------------------------------------------------------------
// Edge pass 3: agg[dst] += alpha * h[src].  One wave32 per edge, each lane
// owns 4 of the 128 channels (float4 gather + 4 native f32 atomic adds).
// ---------------------------------------------------------------------------
__global__ void k_edge_agg(const int* __restrict__ ei, const float* __restrict__ as_,
                           const float* __restrict__ ad_, const float* __restrict__ m_,
                           const float* __restrict__ s_, const float* __restrict__ h,
                           float* __restrict__ agg, int E, int Etot)
{
    const int wid  = blockIdx.x * (blockDim.x >> 5) + (threadIdx.x >> 5);
    const int lane = threadIdx.x & 31;
    if (wid >= Etot) return;
    int src, dst;
    edge_endpoints(ei, wid, E, src, dst);

    const int c    = lane * 4;               // channel base for this lane
    const int head = c >> 6;                 // 64 channels per head
    const float v = leaky(as_[src * HEADS + head] + ad_[dst * HEADS + head]);
    const float alpha = __expf(v - m_[dst * HEADS + head]) /
                        (s_[dst * HEADS + head] + 1e-16f);

    const float4 hv = *(const float4*)(h + (size_t)src * F + c);
    float* __restrict__ ap = agg + (size_t)dst * F + c;
    atomicAdd(ap + 0, hv.x * alpha);
    atomicAdd(ap + 1, hv.y * alpha);
    atomicAdd(ap + 2, hv.z * alpha);
    atomicAdd(ap + 3, hv.w * alpha);
}

// ---------------------------------------------------------------------------
// In-place bias + ELU over the aggregation buffer
// ---------------------------------------------------------------------------
__global__ void k_bias_elu(float* __restrict__ a, const float* __restrict__ bias,
                           int total)
{
    const int tid = blockIdx.x * blockDim.x + threadIdx.x;
    if (tid >= total) return;
    const float v = a[tid] + bias[tid & (F - 1)];
    a[tid] = v > 0.f ? v : (__expf(v) - 1.f);
}

// ---------------------------------------------------------------------------
// Final head: out[n] = <h[n,:], Wr> + br.  One wave per node, shfl reduce.
// ---------------------------------------------------------------------------
__global__ void k_final(const float* __restrict__ h, const float* __restrict__ Wr,
                        const float* __restrict__ br, float* __restrict__ out, int n)
{
    const int wid  = blockIdx.x * (blockDim.x >> 5) + (threadIdx.x >> 5);
    const int lane = threadIdx.x & 31;
    if (wid >= n) return;
    const int c = lane * 4;
    const float4 hv = *(const float4*)(h + (size_t)wid * F + c);
    const float4 wv = *(const float4*)(Wr + c);
    float p = hv.x * wv.x + hv.y * wv.y + hv.z * wv.z + hv.w * wv.w;
    #pragma unroll
    for (int off = 16; off > 0; off >>= 1) p += __shfl_xor(p, off, 32);
    if (lane == 0) out[wid] = p + br[0];
}

// ---------------------------------------------------------------------------
extern "C" void kernel_launch(void* const* d_in, const int* in_sizes, int n_in,
                              void* d_out, int out_size, void* d_ws, size_t ws_size,
                              hipStream_t stream)
{
    const float* x   = (const float*)d_in[0];
    const int*   ei  = (const int*)  d_in[1];
    const float* W1  = (const float*)d_in[2];
    const float* as1 = (const float*)d_in[3];
    const float* ad1 = (const float*)d_in[4];
    const float* b1  = (const float*)d_in[5];
    const float* W2  = (const float*)d_in[6];
    const float* as2 = (const float*)d_in[7];
    const float* ad2 = (const float*)d_in[8];
    const float* b2  = (const float*)d_in[9];
    const float* Wr  = (const float*)d_in[10];
    const float* br  = (const float*)d_in[11];
    float* out = (float*)d_out;

    const int N    = in_sizes[0] / IN_DIM;
    const int E    = in_sizes[1] / 2;
    const int Etot = E + N;                       // reference appends self-loops

    // Workspace layout (floats): ~26.4M floats = ~106 MB
    float* hlin = (float*)d_ws;                   // [N*F] GEMM output h
    float* agg  = hlin + (size_t)N * F;           // [N*F] aggregation / activation
    float* as_  = agg  + (size_t)N * F;           // [N*HEADS]
    float* ad_  = as_  + (size_t)N * HEADS;       // [N*HEADS]
    float* m_   = ad_  + (size_t)N * HEADS;       // [N*HEADS] segment max
    float* s_   = m_   + (size_t)N * HEADS;       // [N*HEADS] segment sum

    const int Mtiles = (N + 15) / 16;
    const int gGemm  = (Mtiles * 4 + 7) / 8;      // 4 waves per M tile, 8 waves/block
    const int gInit  = (N * F + 255) / 256;
    const int gAl    = (N * HEADS + 255) / 256;
    const int gEdge  = (Etot * HEADS + 255) / 256;
    const int gAgg   = (Etot + 7) / 8;            // 8 waves per 256-thread block

    auto run_layer = [&](const float* feat_in, const float* W, const float* av,
                         const float* dv, const float* bias) {
        k_gemm128 <<<gGemm,  256, 0, stream>>>(feat_in, W, hlin, N, Mtiles);
        k_alpha   <<<gAl,    256, 0, stream>>>(hlin, av, dv, as_, ad_, N);
        k_init    <<<gInit,  256, 0, stream>>>(agg, m_, s_, N);
        k_edge_max<<<gEdge,  256, 0, stream>>>(ei, as_, ad_, m_, E, Etot);
        k_edge_sum<<<gEdge,  256, 0, stream>>>(ei, as_, ad_, m_, s_, E, Etot);
        k_edge_agg<<<gAgg,   256, 0, stream>>>(ei, as_, ad_, m_, s_, hlin, agg, E, Etot);
        k_bias_elu<<<gInit,  256, 0, stream>>>(agg, bias, N * F);
    };

    run_layer(x,   W1, as1, ad1, b1);             // layer 1 (input = x)
    run_layer(agg, W2, as2, ad2, b2);             // layer 2 (input = activations)
    k_final<<<(N + 7) / 8, 256, 0, stream>>>(agg, Wr, br, out, N);
}